// GINConv_87806311399693
// MI455X (gfx1250) — compile-verified
//
#include <hip/hip_runtime.h>

typedef __attribute__((ext_vector_type(2))) float v2f;
typedef __attribute__((ext_vector_type(8))) float v8f;

#define N_NODES 50000
#define DDIM    256
#define N_EDGES 1600000

// ---------------- zero workspace ----------------
__global__ void gin_zero_kernel(float4* __restrict__ p, int n4) {
    int i = blockIdx.x * blockDim.x + threadIdx.x;
    if (i < n4) p[i] = make_float4(0.f, 0.f, 0.f, 0.f);
}

// ---------------- edge scatter-add --------------
// 64 threads per edge, each handles one float4 chunk of the 256-wide feature.
// x (51.2MB) fits in the 192MB L2, so the random gather is L2-resident; the
// adds are f32 atomics resolved at L2.
__global__ void gin_scatter_kernel(const float* __restrict__ x,
                                   const int* __restrict__ ei,
                                   float* __restrict__ agg) {
    int gid = blockIdx.x * blockDim.x + threadIdx.x;   // < 1.6M * 64
    int e = gid >> 6;
    int c = gid & 63;
    if (e >= N_EDGES) return;
    int src = ei[e];
    int dst = ei[N_EDGES + e];
    const float4 v = *(const float4*)(x + (size_t)src * DDIM + c * 4);
    float* o = agg + (size_t)dst * DDIM + c * 4;
    atomicAdd(o + 0, v.x);
    atomicAdd(o + 1, v.y);
    atomicAdd(o + 2, v.z);
    atomicAdd(o + 3, v.w);
}

// ---------------- fused WMMA GEMM ---------------
// C[M,256] = act(Ain @ W + bias),  Ain = FUSE_EPS ? (1+eps)*A + A2 : A
// Block: 256 threads = 8 waves. Block tile 128(M) x 64(N); each wave: 16x64.
// K chunked by 32 through LDS; inner matrix op: v_wmma_f32_16x16x4_f32.
// B is stored K-pair interleaved so each B fragment {W[k][c], W[k+1][c]} is a
// single contiguous ds_load_b64; kp-row stride 160 floats == 32 banks so the
// two lane halves (kp vs kp+1) hit disjoint bank halves.
template <bool FUSE_EPS, bool RELU>
__global__ void gin_gemm_kernel(const float* __restrict__ A,
                                const float* __restrict__ A2,
                                const float* __restrict__ eps,
                                const float* __restrict__ W,
                                const float* __restrict__ bias,
                                float* __restrict__ C) {
    __shared__ float As[128][36];   // stride 36: even (b64-aligned), conflict-free
    __shared__ float Bs[16][160];   // [k/2][col*2 + (k&1)], stride 160 = 32 mod 64 banks

    const int m0   = blockIdx.x * 128;
    const int n0   = blockIdx.y * 64;
    const int lane = threadIdx.x & 31;
    const int wv   = threadIdx.x >> 5;
    const int l15  = lane & 15;
    const int hi   = lane >> 4;                 // 0: lanes 0-15, 1: lanes 16-31
    const bool wvalid = (m0 + wv * 16) < N_NODES;

    const float epsv = FUSE_EPS ? (1.0f + eps[0]) : 0.0f;

    v8f acc[4] = {};

    for (int kc = 0; kc < DDIM; kc += 32) {
        // ---- cooperative A tile load: 128 rows x 32 cols (1024 float4) ----
        #pragma unroll
        for (int it = 0; it < 4; ++it) {
            int idx  = threadIdx.x + it * 256;  // 0..1023
            int row  = idx >> 3;
            int c4   = idx & 7;
            int grow = m0 + row;
            float4 v = make_float4(0.f, 0.f, 0.f, 0.f);
            if (grow < N_NODES) {
                v = ((const float4*)(A + (size_t)grow * DDIM + kc))[c4];
                if (FUSE_EPS) {
                    float4 g = ((const float4*)(A2 + (size_t)grow * DDIM + kc))[c4];
                    v.x = epsv * v.x + g.x;
                    v.y = epsv * v.y + g.y;
                    v.z = epsv * v.z + g.z;
                    v.w = epsv * v.w + g.w;
                }
            }
            float* d = &As[row][c4 * 4];        // byte offset 144*row+16*c4 (8B aligned)
            ((float2*)d)[0] = make_float2(v.x, v.y);
            ((float2*)d)[1] = make_float2(v.z, v.w);
        }
        // ---- cooperative B tile load: 32 K-rows x 64 cols, K-pair interleaved ----
        #pragma unroll
        for (int it = 0; it < 2; ++it) {
            int idx = threadIdx.x + it * 256;   // 0..511
            int kr  = idx >> 4;                 // K row within chunk
            int c4  = idx & 15;                 // float4 column group
            float4 v = ((const float4*)(W + (size_t)(kc + kr) * DDIM + n0))[c4];
            float* d = &Bs[kr >> 1][c4 * 8 + (kr & 1)];
            d[0] = v.x; d[2] = v.y; d[4] = v.z; d[6] = v.w;
        }
        __syncthreads();

        if (wvalid) {                           // wave-uniform: EXEC all-1s for WMMA
            #pragma unroll
            for (int kk = 0; kk < 32; kk += 4) {
                int kb = kk + hi * 2;
                // A frag (16x4 f32): lane<16 -> {K=kk,kk+1}, lane>=16 -> {K=kk+2,kk+3}
                v2f af = *(const v2f*)&As[wv * 16 + l15][kb];
                const float* brow = &Bs[kb >> 1][0];
                #pragma unroll
                for (int nt = 0; nt < 4; ++nt) {
                    // B frag: contiguous pair {W[kb][c], W[kb+1][c]} -> ds_load_b64
                    v2f bf = *(const v2f*)(brow + (nt * 16 + l15) * 2);
                    acc[nt] = __builtin_amdgcn_wmma_f32_16x16x4_f32(
                        false, af, false, bf, (short)0, acc[nt], false, false);
                }
            }
        }
        __syncthreads();
    }

    if (wvalid) {
        // C/D layout: lane<16 -> M = rbase+r, N = n0+nt*16+l15; lane>=16 -> M += 8
        int rbase = m0 + wv * 16 + hi * 8;
        #pragma unroll
        for (int nt = 0; nt < 4; ++nt) {
            int col = n0 + nt * 16 + l15;
            float bv = bias[col];
            #pragma unroll
            for (int r = 0; r < 8; ++r) {
                float v = acc[nt][r] + bv;
                if (RELU) v = fmaxf(v, 0.0f);
                C[(size_t)(rbase + r) * DDIM + col] = v;
            }
        }
    }
}

extern "C" void kernel_launch(void* const* d_in, const int* in_sizes, int n_in,
                              void* d_out, int out_size, void* d_ws, size_t ws_size,
                              hipStream_t stream) {
    const float* x   = (const float*)d_in[0];
    const int*   ei  = (const int*)d_in[1];
    // d_in[2] = edge_attr (unused by forward)
    const float* W1  = (const float*)d_in[3];
    const float* b1  = (const float*)d_in[4];
    const float* W2  = (const float*)d_in[5];
    const float* b2  = (const float*)d_in[6];
    const float* eps = (const float*)d_in[7];
    float* out = (float*)d_out;

    float* agg = (float*)d_ws;
    float* h   = agg + (size_t)N_NODES * DDIM;

    // 1) zero agg
    int n4 = N_NODES * DDIM / 4;
    gin_zero_kernel<<<(n4 + 255) / 256, 256, 0, stream>>>((float4*)agg, n4);

    // 2) scatter-sum over edges
    long long tot = (long long)N_EDGES * 64;
    gin_scatter_kernel<<<(unsigned)((tot + 255) / 256), 256, 0, stream>>>(x, ei, agg);

    // 3) h = relu(((1+eps)x + agg) @ W1 + b1)   (eps-combine fused into A load)
    dim3 grid((N_NODES + 127) / 128, DDIM / 64);
    gin_gemm_kernel<true, true><<<grid, 256, 0, stream>>>(x, agg, eps, W1, b1, h);

    // 4) out = h @ W2 + b2
    gin_gemm_kernel<false, false><<<grid, 256, 0, stream>>>(h, nullptr, nullptr, W2, b2, out);
}